// DeconvWithPruning_40312563040282
// MI455X (gfx1250) — compile-verified
//
#include <hip/hip_runtime.h>

typedef float v2f __attribute__((ext_vector_type(2)));
typedef float v8f __attribute__((ext_vector_type(8)));

#define RAVEL_STEP 514  // 2*GRID+2 from the reference

__device__ __forceinline__ int ravel3(int x, int y, int z) {
    return (x * RAVEL_STEP + y) * RAVEL_STEP + z;
}

__device__ __forceinline__ int lower_bound_i32(const int* __restrict__ keys, int n, int key) {
    int lo = 0, hi = n;
    while (lo < hi) {
        int mid = (lo + hi) >> 1;
        if (keys[mid] < key) lo = mid + 1; else hi = mid;
    }
    return lo;
}

// ---- kernel 1: build sorted key arrays for out_coords / ref_coords -------------
__global__ void dp_build_keys(const int* __restrict__ out_coords,
                              const int* __restrict__ ref_coords,
                              int* __restrict__ out_keys,
                              int* __restrict__ ref_keys,
                              int n_out, int n_ref) {
    int i = blockIdx.x * blockDim.x + threadIdx.x;
    if (i < n_out)
        out_keys[i] = ravel3(out_coords[i * 4 + 1], out_coords[i * 4 + 2], out_coords[i * 4 + 3]);
    if (i < n_ref)
        ref_keys[i] = ravel3(ref_coords[i * 4 + 1], ref_coords[i * 4 + 2], ref_coords[i * 4 + 3]);
}

// ---- kernel 1b: transpose W[k][64][32] -> Wt[k][32][64] so B fragments are
// contiguous v2f along K (enables global_load_b64 in the WMMA inner loop) -------
__global__ void dp_transpose_w(const float* __restrict__ W, float* __restrict__ Wt) {
    int i = blockIdx.x * blockDim.x + threadIdx.x;  // 27*2048 elements
    if (i >= 27 * 2048) return;
    int k = i >> 11;
    int r = i & 2047;
    int n = r >> 6;       // output channel 0..31
    int c = r & 63;       // input channel  0..63
    Wt[i] = W[k * 2048 + c * 32 + n];
}

// ---- kernel 2: zero the accumulation buffer (d_out is poisoned to 0xAA) --------
__global__ void dp_zero(float4* __restrict__ p, long n4) {
    long i = blockIdx.x * (long)blockDim.x + threadIdx.x;
    if (i < n4) p[i] = make_float4(0.f, 0.f, 0.f, 0.f);
}

// ---- kernel 3: WMMA gather-GEMM-scatter ----------------------------------------
// One wave32 per 16-row input tile. A = x tile [16x64] fp32, B = Wt[k] (K-major),
// accumulated with V_WMMA_F32_16X16X4_F32 over 16 K-steps, two 16-wide N halves.
__global__ __launch_bounds__(256)
void dp_gemm_scatter(const float* __restrict__ x,
                     const float* __restrict__ Wt,
                     const int* __restrict__ in_coords,
                     const int* __restrict__ out_keys,
                     float* __restrict__ feats,
                     int n_in, int n_out) {
    const int lane = threadIdx.x & 31;
    const int wave = threadIdx.x >> 5;
    const int tile = blockIdx.x * 8 + wave;
    const int row0 = tile * 16;
    if (row0 >= n_in) return;  // uniform per wave

    const int m    = lane & 15;          // A row (M) this lane carries
    const int koff = (lane >> 4) * 2;    // K sub-offset per ISA 16x4 fp32 A layout
    const int rowm = row0 + m;
    const bool valid = rowm < n_in;
    const int rowc = valid ? rowm : (n_in - 1);

    // Preload A fragments: a[kk] covers global K = 4*kk + koff + {0,1}
    v2f a[16];
    const float* xr = x + (size_t)rowc * 64;
    #pragma unroll
    for (int kk = 0; kk < 16; ++kk) {
        v2f av = *(const v2f*)(xr + 4 * kk + koff);
        a[kk].x = valid ? av.x : 0.0f;
        a[kk].y = valid ? av.y : 0.0f;
    }

    // This lane's input coordinate (lanes 16..31 duplicate lanes 0..15)
    const int cx = in_coords[rowc * 4 + 1];
    const int cy = in_coords[rowc * 4 + 2];
    const int cz = in_coords[rowc * 4 + 3];

    const int n0 = lane & 15;  // B/D column within a 16-wide half
    const float* __restrict__ brow0 = Wt + (size_t)n0 * 64 + koff;         // N half 0
    const float* __restrict__ brow1 = Wt + (size_t)(n0 + 16) * 64 + koff;  // N half 1

    for (int k = 0; k < 27; ++k) {
        const int ox = k / 9 - 1, oy = (k / 3) % 3 - 1, oz = k % 3 - 1;
        // scatter row for input row (lane&15) at this kernel offset
        const int key = ravel3(cx + ox, cy + oy, cz + oz);
        const int idx = lower_bound_i32(out_keys, n_out, key);

        const size_t kb = (size_t)k * 2048;
        v8f c0 = {}; v8f c1 = {};
        #pragma unroll
        for (int kk = 0; kk < 16; ++kk) {
            v2f b0 = *(const v2f*)(brow0 + kb + 4 * kk);  // Wt[k][n0][kr..kr+1]
            v2f b1 = *(const v2f*)(brow1 + kb + 4 * kk);  // Wt[k][n0+16][kr..kr+1]
            c0 = __builtin_amdgcn_wmma_f32_16x16x4_f32(false, a[kk], false, b0,
                                                       (short)0, c0, false, false);
            c1 = __builtin_amdgcn_wmma_f32_16x16x4_f32(false, a[kk], false, b1,
                                                       (short)0, c1, false, false);
        }

        // D layout: vgpr v, lane L -> row v + 8*(L>>4), col L&15
        #pragma unroll
        for (int v = 0; v < 8; ++v) {
            const int mrow = v + ((lane >> 4) << 3);
            const int tgt  = __shfl(idx, mrow, 32);
            if (row0 + mrow < n_in) {
                float* p = feats + (size_t)tgt * 32 + n0;
                unsafeAtomicAdd(p, c0[v]);
                unsafeAtomicAdd(p + 16, c1[v]);
            }
        }
    }
}

// ---- kernel 4: bias + pruning mask, emit feats and float mask ------------------
__global__ __launch_bounds__(256)
void dp_finalize(const float* __restrict__ bias,
                 const int* __restrict__ out_keys,
                 const int* __restrict__ ref_keys,
                 float* __restrict__ feats,
                 float* __restrict__ mask_out,
                 int n_out, int n_ref) {
    int row = blockIdx.x * blockDim.x + threadIdx.x;
    if (row >= n_out) return;
    const int key = out_keys[row];
    const int pos = lower_bound_i32(ref_keys, n_ref, key);
    const bool keep = (pos < n_ref) && (ref_keys[pos] == key);
    const float fm = keep ? 1.0f : 0.0f;
    float* p = feats + (size_t)row * 32;
    #pragma unroll
    for (int c = 0; c < 32; ++c) p[c] = (p[c] + bias[c]) * fm;
    mask_out[row] = fm;
}

extern "C" void kernel_launch(void* const* d_in, const int* in_sizes, int n_in_arrs,
                              void* d_out, int out_size, void* d_ws, size_t ws_size,
                              hipStream_t stream) {
    const float* x    = (const float*)d_in[0];
    const float* W    = (const float*)d_in[1];
    const float* bias = (const float*)d_in[2];
    const int*   inc  = (const int*)d_in[3];
    const int*   outc = (const int*)d_in[4];
    const int*   refc = (const int*)d_in[5];

    const int n_in  = in_sizes[0] / 64;  // C_IN = 64
    const int n_out = in_sizes[4] / 4;
    const int n_ref = in_sizes[5] / 4;

    float* feats = (float*)d_out;
    float* mask  = feats + (size_t)n_out * 32;

    int*   out_keys = (int*)d_ws;
    int*   ref_keys = out_keys + n_out;
    float* Wt       = (float*)(ref_keys + n_ref);   // 27*2048 floats (221 KB)

    {
        int n = n_out > n_ref ? n_out : n_ref;
        dp_build_keys<<<(n + 255) / 256, 256, 0, stream>>>(outc, refc, out_keys, ref_keys,
                                                           n_out, n_ref);
    }
    dp_transpose_w<<<(27 * 2048 + 255) / 256, 256, 0, stream>>>(W, Wt);
    {
        long n4 = (long)n_out * 8;  // n_out*32 floats as float4
        dp_zero<<<(int)((n4 + 255) / 256), 256, 0, stream>>>((float4*)feats, n4);
    }
    {
        int tiles  = (n_in + 15) / 16;
        int blocks = (tiles + 7) / 8;  // 8 waves (tiles) per block
        dp_gemm_scatter<<<blocks, 256, 0, stream>>>(x, Wt, inc, out_keys, feats, n_in, n_out);
    }
    dp_finalize<<<(n_out + 255) / 256, 256, 0, stream>>>(bias, out_keys, ref_keys,
                                                         feats, mask, n_out, n_ref);
}